// ProjectionHelper_24498493456678
// MI455X (gfx1250) — compile-verified
//
#include <hip/hip_runtime.h>
#include <cstdint>
#include <cstddef>

#define IMG_DIM   256
#define IMG_HW    (256 * 256)
#define BLOCK_PTS 256

// ---------------------------------------------------------------------------
// Kernel 1: per-batch projection matrix  P(3x4) = intr(3x3) @ inv(ext)[0:3, :]
// Gauss-Jordan with partial pivoting; one thread per batch (B = 8).
// Runtime-negligible (one 32-thread block).
// ---------------------------------------------------------------------------
__global__ void ph_build_proj_kernel(const float* __restrict__ ext,
                                     const float* __restrict__ intr,
                                     float* __restrict__ P, int Bn) {
  const int b = threadIdx.x;
  if (b >= Bn) return;

  float m[4][8];
  for (int r = 0; r < 4; ++r)
    for (int c = 0; c < 4; ++c) {
      m[r][c]     = ext[b * 16 + r * 4 + c];
      m[r][4 + c] = (r == c) ? 1.0f : 0.0f;
    }

  for (int col = 0; col < 4; ++col) {
    int piv = col;
    float best = fabsf(m[col][col]);
    for (int r = col + 1; r < 4; ++r) {
      float a = fabsf(m[r][col]);
      if (a > best) { best = a; piv = r; }
    }
    if (piv != col)
      for (int c = 0; c < 8; ++c) {
        float t = m[col][c]; m[col][c] = m[piv][c]; m[piv][c] = t;
      }
    float inv = 1.0f / m[col][col];
    for (int c = 0; c < 8; ++c) m[col][c] *= inv;
    for (int r = 0; r < 4; ++r) {
      if (r == col) continue;
      float f = m[r][col];
      for (int c = 0; c < 8; ++c) m[r][c] = fmaf(-f, m[col][c], m[r][c]);
    }
  }

  // P = intr @ invE[:3,:4]
  for (int r = 0; r < 3; ++r)
    for (int c = 0; c < 4; ++c) {
      float s = 0.0f;
      for (int j = 0; j < 3; ++j) s = fmaf(intr[r * 3 + j], m[j][4 + c], s);
      P[b * 12 + r * 4 + c] = s;
    }
}

// ---------------------------------------------------------------------------
// Kernel 2: project + bilinear gather.
// One thread per point. Point coords staged into LDS via the gfx1250 async
// global->LDS path (coalesced b32 lanes), fenced with s_wait_asynccnt.
// Gather loads use default (RT) caching so the ~16 MB hot quadrant stays in
// the 192 MB L2; output stores are non-temporal (102 MB write-once stream).
// KC_T > 0: compile-time channel count (immediate-offset gather loads).
// ---------------------------------------------------------------------------
template <int KC_T>
__global__ __launch_bounds__(BLOCK_PTS)
void ph_project_gather_kernel(const float* __restrict__ pts,
                              const float* __restrict__ feat,
                              const float* __restrict__ P,
                              float* __restrict__ out,
                              int N, int Kc_rt) {
  __shared__ float spts[BLOCK_PTS * 3];

  const int b  = blockIdx.y;
  const int n0 = blockIdx.x * BLOCK_PTS;
  const int t  = threadIdx.x;
  const int Kc = (KC_T > 0) ? KC_T : Kc_rt;

  // ---- async stage: BLOCK_PTS*3 contiguous floats -> LDS --------------------
  const float* src   = pts + ((size_t)b * N + n0) * 3;
  const int remain_f = (N - n0 < BLOCK_PTS ? N - n0 : BLOCK_PTS) * 3;

  #pragma unroll
  for (int i = 0; i < 3; ++i) {
    const int e = t + i * BLOCK_PTS;
    if (e < remain_f) {
      const uint32_t lds_addr = (uint32_t)(uintptr_t)(&spts[e]);
      const uint64_t gaddr    = (uint64_t)(uintptr_t)(src + e);
      asm volatile("global_load_async_to_lds_b32 %0, %1, off"
                   :: "v"(lds_addr), "v"(gaddr)
                   : "memory");
    }
  }
#if __has_builtin(__builtin_amdgcn_s_wait_asynccnt)
  __builtin_amdgcn_s_wait_asynccnt(0);
#else
  asm volatile("s_wait_asynccnt 0" ::: "memory");
#endif
  __syncthreads();

  // ---- projection (P loads are uniform -> scalar loads) ---------------------
  const int  n      = n0 + t;
  const bool active = n < N;

  float x = 0.0f, y = 0.0f, z = 0.0f;
  if (active) {
    x = spts[t * 3 + 0];
    y = spts[t * 3 + 1];
    z = spts[t * 3 + 2];
  }

  const float* Pb = P + b * 12;
  const float ix = fmaf(Pb[0], x, fmaf(Pb[1], y, fmaf(Pb[2],  z, Pb[3])));
  const float iy = fmaf(Pb[4], x, fmaf(Pb[5], y, fmaf(Pb[6],  z, Pb[7])));
  const float iz = fmaf(Pb[8], x, fmaf(Pb[9], y, fmaf(Pb[10], z, Pb[11])));

  const float rz = 1.0f / iz;
  const float u  = ix * rz;
  const float v  = iy * rz;

  const bool valid = (u >= 0.0f) & (v >= 0.0f) &
                     (u < (float)IMG_DIM) & (v < (float)IMG_DIM);

  const float uc = fminf(fmaxf(u, 1.0f), (float)(IMG_DIM - 2));
  const float vc = fminf(fmaxf(v, 1.0f), (float)(IMG_DIM - 2));
  const float fx = floorf(uc);
  const float fy = floorf(vc);
  const int   x0 = (int)fx;
  const int   y0 = (int)fy;

  const float tfx = uc - fx;
  const float tfy = vc - fy;
  const float tcx = (fx + 1.0f) - uc;   // matches reference's cli - imc
  const float tcy = (fy + 1.0f) - vc;

  float w00 = tcy * tcx;
  float w01 = tcy * tfx;
  float w10 = tfy * tcx;
  float w11 = tfy * tfx;
  if (!valid) { w00 = 0.0f; w01 = 0.0f; w10 = 0.0f; w11 = 0.0f; }

  const int idx = y0 * IMG_DIM + x0;
  const float* __restrict__ fb = feat + (size_t)b * Kc * IMG_HW + idx;
  float* __restrict__ ob       = out  + (size_t)b * Kc * N + n;

  if (active) {
    #pragma unroll 8
    for (int k = 0; k < Kc; ++k) {
      const float* fk = fb + (size_t)k * IMG_HW;   // const stride -> imm offsets
      const float f00 = fk[0];
      const float f01 = fk[1];
      const float f10 = fk[IMG_DIM];
      const float f11 = fk[IMG_DIM + 1];
      float r = f00 * w00;
      r = fmaf(f01, w01, r);
      r = fmaf(f10, w10, r);
      r = fmaf(f11, w11, r);
      __builtin_nontemporal_store(r, &ob[(size_t)k * N]);  // NT: don't evict L2 hot set
    }
  }
}

// ---------------------------------------------------------------------------
// Host launcher
// ---------------------------------------------------------------------------
extern "C" void kernel_launch(void* const* d_in, const int* in_sizes, int n_in,
                              void* d_out, int out_size, void* d_ws, size_t ws_size,
                              hipStream_t stream) {
  const float* pts  = (const float*)d_in[0];   // (B, N, 3)
  const float* feat = (const float*)d_in[1];   // (B, K, 256, 256)
  const float* ext  = (const float*)d_in[2];   // (B, 4, 4)
  const float* intr = (const float*)d_in[3];   // (3, 3)
  float* out        = (float*)d_out;           // (B, K, N)

  const int B  = in_sizes[2] / 16;
  const int N  = in_sizes[0] / (3 * B);
  const int Kc = in_sizes[1] / (B * IMG_HW);

  float* P = (float*)d_ws;                     // B * 12 floats

  ph_build_proj_kernel<<<1, 32, 0, stream>>>(ext, intr, P, B);

  dim3 grid((N + BLOCK_PTS - 1) / BLOCK_PTS, B, 1);
  if (Kc == 32) {
    ph_project_gather_kernel<32><<<grid, BLOCK_PTS, 0, stream>>>(pts, feat, P, out, N, Kc);
  } else {
    ph_project_gather_kernel<0><<<grid, BLOCK_PTS, 0, stream>>>(pts, feat, P, out, N, Kc);
  }
}